// CVRPLoss_80650895884983
// MI455X (gfx1250) — compile-verified
//
#include <hip/hip_runtime.h>
#include <hip/hip_bf16.h>
#include <cmath>

typedef __attribute__((ext_vector_type(2))) float v2f;
typedef __attribute__((ext_vector_type(8))) float v8f;

// Accumulator slot indices (placed in d_ws after in_sums/out_sums)
#define ACC_FOCAL    0
#define ACC_COVIN    1
#define ACC_COVOUT   2
#define ACC_TOUR     3
#define ACC_DEMAND   4
#define ACC_NODESUM  5
#define ACC_MASKCNT  6
#define N_ACC        8

// ---------------------------------------------------------------------------
// Full-precision 32-lane wave sum using V_WMMA_F32_16X16X4_F32.
// A (16x4 f32, 2 VGPRs): VGPR0 = per-lane value v, VGPR1 = 0.
// B (4x16) = all ones  ->  D[m][n] = v(m) + v(m+16) for every column n.
// Lane n (<16) holds rows 0-7 of column n in its 8 D VGPRs; lane n+16 holds
// rows 8-15. Per-lane sum of the 8 D registers + one xor-16 bpermute yields
// the full 32-lane sum in every lane, entirely in f32.
// Requires EXEC == all ones (call only from converged code).
// ---------------------------------------------------------------------------
__device__ __forceinline__ float wave_sum32(float v) {
    v2f a; a[0] = v;    a[1] = 0.0f;
    v2f b; b[0] = 1.0f; b[1] = 1.0f;
    v8f c = {};
    c = __builtin_amdgcn_wmma_f32_16x16x4_f32(
        /*neg_a=*/false, a, /*neg_b=*/false, b,
        /*c_mod=*/(short)0, c, /*reuse_a=*/false, /*reuse_b=*/false);
    float s = ((c[0] + c[1]) + (c[2] + c[3])) + ((c[4] + c[5]) + (c[6] + c[7]));
    s += __shfl_xor(s, 16, 32);   // combine rows 0-7 half with rows 8-15 half
    return s;                     // identical in all 32 lanes
}

// ---------------------------------------------------------------------------
__global__ void k_init(float* ws, int n) {
    int i = blockIdx.x * blockDim.x + threadIdx.x;
    if (i < n) ws[i] = 0.0f;
}

// ---------------------------------------------------------------------------
// Edge pass: sigmoid + scatter-add (segment sums) + focal-loss partial.
// Streams ~156 MB once; HBM-bound (~7 us floor at 23.3 TB/s). Scatter
// targets (2 x 400 KB f32) stay resident in the 192 MB L2, so the
// global_atomic_add_f32 traffic resolves in-cache.
// Vectorized path: 4 edges/thread/iter via B128/B64 loads (wave32 issues
// 512 B contiguous per load instruction).
// ---------------------------------------------------------------------------
__device__ __forceinline__ float edge_term(float l, float y,
                                           float* __restrict__ in_sums,
                                           float* __restrict__ out_sums,
                                           int s, int d) {
    float p = 1.0f / (1.0f + expf(-l));                  // sigmoid
    atomicAdd(&in_sums[d],  p);                          // global_atomic_add_f32
    atomicAdd(&out_sums[s], p);
    // focal loss element (alpha=0.25, gamma=2)
    float bce     = fmaxf(l, 0.0f) - l * y + log1pf(expf(-fabsf(l)));
    float p_t     = p * y + (1.0f - p) * (1.0f - y);
    float alpha_t = 0.25f * y + 0.75f * (1.0f - y);
    float om      = 1.0f - p_t;
    return alpha_t * om * om * bce;
}

__global__ void k_edges(const float* __restrict__ ep,
                        const float* __restrict__ ye,
                        const long long* __restrict__ ei,   // [2, n_edges] i64
                        float* __restrict__ in_sums,
                        float* __restrict__ out_sums,
                        float* __restrict__ acc,
                        int n_edges) {
    const int tid    = blockIdx.x * blockDim.x + threadIdx.x;
    const int stride = gridDim.x * blockDim.x;

    float focal = 0.0f;

    if ((n_edges & 3) == 0) {
        // ---- vectorized path: groups of 4 edges, B128 loads ----
        const int nvec = n_edges >> 2;
        const float4*    ep4 = (const float4*)ep;
        const float4*    ye4 = (const float4*)ye;
        const longlong2* es2 = (const longlong2*)ei;              // src half
        const longlong2* ed2 = (const longlong2*)(ei + n_edges);  // dst half

        for (int g = tid; g < nvec; g += stride) {
            if (g + stride < nvec) {
                // WGP-scope prefetch of next iteration's cachelines
                __builtin_prefetch(ep4 + g + stride, 0, 3);   // global_prefetch_b8
                __builtin_prefetch(ye4 + g + stride, 0, 3);
                __builtin_prefetch(es2 + 2 * (g + stride), 0, 3);
                __builtin_prefetch(ed2 + 2 * (g + stride), 0, 3);
            }
            float4    l4  = ep4[g];
            float4    y4  = ye4[g];
            longlong2 s01 = es2[2 * g];
            longlong2 s23 = es2[2 * g + 1];
            longlong2 d01 = ed2[2 * g];
            longlong2 d23 = ed2[2 * g + 1];

            focal += edge_term(l4.x, y4.x, in_sums, out_sums, (int)s01.x, (int)d01.x);
            focal += edge_term(l4.y, y4.y, in_sums, out_sums, (int)s01.y, (int)d01.y);
            focal += edge_term(l4.z, y4.z, in_sums, out_sums, (int)s23.x, (int)d23.x);
            focal += edge_term(l4.w, y4.w, in_sums, out_sums, (int)s23.y, (int)d23.y);
        }
    } else {
        // ---- scalar fallback (alignment-safe for any n_edges) ----
        for (int e = tid; e < n_edges; e += stride) {
            focal += edge_term(ep[e], ye[e], in_sums, out_sums,
                               (int)ei[e], (int)ei[n_edges + e]);
        }
    }

    // converged here: EXEC all ones -> WMMA wave reduction is legal
    float wsum = wave_sum32(focal);
    if ((threadIdx.x & 31) == 0) atomicAdd(&acc[ACC_FOCAL], wsum);
}

// ---------------------------------------------------------------------------
// Node pass: coverage / tour / demand / masked node-MSE partial sums.
// Only ~2.8 MB of traffic; x rows are natural float4 (B128) loads.
// ---------------------------------------------------------------------------
__global__ void k_nodes(const float* __restrict__ npred,
                        const float* __restrict__ x,       // [n_nodes, 4]
                        const float* __restrict__ yn,
                        const float* __restrict__ in_sums,
                        const float* __restrict__ out_sums,
                        float* __restrict__ acc,
                        int n_nodes) {
    const int tid    = blockIdx.x * blockDim.x + threadIdx.x;
    const int stride = gridDim.x * blockDim.x;
    const float4* x4 = (const float4*)x;

    float cov_in = 0.0f, cov_out = 0.0f, tour = 0.0f;
    float demand = 0.0f, nsum = 0.0f, cnt = 0.0f;

    for (int i = tid; i < n_nodes; i += stride) {
        float is = in_sums[i];
        float os = out_sums[i];
        float di = is - 1.0f;
        float dv = os - 1.0f;
        if (i >= 1) {
            cov_in  += di * di;
            cov_out += dv * dv;
            demand  += x4[i].z;          // x[:,2], one B128 row load
        }
        float t = is - os;
        tour += t * t;

        float yv = yn[i];
        float m  = (yv >= 0.0f) ? 1.0f : 0.0f;
        float e  = npred[i] - yv;
        nsum += m * e * e;
        cnt  += m;
    }

    float r;
    r = wave_sum32(cov_in);  if ((threadIdx.x & 31) == 0) atomicAdd(&acc[ACC_COVIN],   r);
    r = wave_sum32(cov_out); if ((threadIdx.x & 31) == 0) atomicAdd(&acc[ACC_COVOUT],  r);
    r = wave_sum32(tour);    if ((threadIdx.x & 31) == 0) atomicAdd(&acc[ACC_TOUR],    r);
    r = wave_sum32(demand);  if ((threadIdx.x & 31) == 0) atomicAdd(&acc[ACC_DEMAND],  r);
    r = wave_sum32(nsum);    if ((threadIdx.x & 31) == 0) atomicAdd(&acc[ACC_NODESUM], r);
    r = wave_sum32(cnt);     if ((threadIdx.x & 31) == 0) atomicAdd(&acc[ACC_MASKCNT], r);
}

// ---------------------------------------------------------------------------
__global__ void k_final(const float* __restrict__ in_sums,
                        const float* __restrict__ out_sums,
                        const float* __restrict__ acc,
                        const float* __restrict__ capacity,
                        float* __restrict__ out,
                        int n_nodes, int n_edges) {
    if (blockIdx.x != 0 || threadIdx.x != 0) return;

    float in0  = in_sums[0];
    float out0 = out_sums[0];

    float denom    = 2.0f * (float)((n_nodes - 1) > 1 ? (n_nodes - 1) : 1);
    float coverage = (acc[ACC_COVIN] + acc[ACC_COVOUT]) / denom;
    float tour     = acc[ACC_TOUR] / (float)n_nodes;
    float dep      = in0 - out0;
    float depot    = dep * dep;

    float cap       = capacity[0];                   // mean of size-1 array
    float expected  = ceilf(acc[ACC_DEMAND] / cap);
    float ct        = out0 - expected;
    float cap_tours = ct * ct;

    float similarity = acc[ACC_FOCAL] / (float)n_edges;
    float cntc       = fmaxf(acc[ACC_MASKCNT], 1.0f);
    float node_loss  = acc[ACC_NODESUM] / cntc;

    out[0] = coverage * 5.0f + tour * 3.0f + depot * 2.0f +
             cap_tours * 1.5f + similarity * 0.3f + node_loss * 0.1f;
}

// ---------------------------------------------------------------------------
extern "C" void kernel_launch(void* const* d_in, const int* in_sizes, int n_in,
                              void* d_out, int out_size, void* d_ws, size_t ws_size,
                              hipStream_t stream) {
    // setup_inputs() order:
    // 0: edge_predictions f32 [E]
    // 1: node_predictions f32 [N]
    // 2: x                f32 [N,4]
    // 3: capacity         f32 [1]
    // 4: y_edges          f32 [E]
    // 5: y_nodes          f32 [N]
    // 6: edge_index       i64 [2,E]
    // 7: num_nodes        (scalar)
    const float*     ep  = (const float*)d_in[0];
    const float*     np_ = (const float*)d_in[1];
    const float*     x   = (const float*)d_in[2];
    const float*     cap = (const float*)d_in[3];
    const float*     ye  = (const float*)d_in[4];
    const float*     yn  = (const float*)d_in[5];
    const long long* ei  = (const long long*)d_in[6];

    const int n_edges = in_sizes[0];
    const int n_nodes = in_sizes[1];

    // d_ws layout: [in_sums (N)] [out_sums (N)] [N_ACC accumulators]
    float* ws       = (float*)d_ws;
    float* in_sums  = ws;
    float* out_sums = ws + n_nodes;
    float* acc      = ws + 2 * n_nodes;
    const int n_zero = 2 * n_nodes + N_ACC;

    const int BLK = 256;  // 8 wave32s

    k_init<<<(n_zero + BLK - 1) / BLK, BLK, 0, stream>>>(ws, n_zero);

    // Vectorized kernel consumes 4 edges/thread/iter: size the grid so each
    // thread runs ~3 iterations (keeps grid-stride prefetch meaningful while
    // still launching ~16K wave32s for latency hiding).
    int work_items = ((n_edges & 3) == 0) ? (n_edges >> 2) : n_edges;
    int blk_e = (work_items + BLK - 1) / BLK;
    if (blk_e > 2048) blk_e = 2048;
    k_edges<<<blk_e, BLK, 0, stream>>>(ep, ye, ei, in_sums, out_sums, acc, n_edges);

    int blk_n = (n_nodes + BLK - 1) / BLK;
    if (blk_n > 1024) blk_n = 1024;
    k_nodes<<<blk_n, BLK, 0, stream>>>(np_, x, yn, in_sums, out_sums, acc, n_nodes);

    k_final<<<1, 32, 0, stream>>>(in_sums, out_sums, acc, cap, (float*)d_out,
                                  n_nodes, n_edges);
}